// LLObjectCondensation_35768487641758
// MI455X (gfx1250) — compile-verified
//
#include <hip/hip_runtime.h>

#define QMIN     0.5f
#define SB       1.0f
#define HUBER    2.0f
#define EDEN     1.0f
#define KMAX     192
#define EPSF     1e-6f
#define NEVENTS  4

typedef float v2f __attribute__((ext_vector_type(2)));
typedef float v8f __attribute__((ext_vector_type(8)));

__device__ __forceinline__ float clipb(float b) {
    return fminf(fmaxf(b, 1e-4f), 1.0f - 1e-4f);
}
__device__ __forceinline__ float qval(float b) {
    float t = atanhf(clipb(b));
    return t * t + QMIN;
}

// ---------------------------------------------------------------- kernel 0
__global__ void k_init(unsigned long long* amax, float* spw, float* spweh,
                       float* nsum, float* ncnt, int BK, int Bv) {
    int i = blockIdx.x * blockDim.x + threadIdx.x;
    if (i < BK) { amax[i] = 0ull; spw[i] = 0.0f; spweh[i] = 0.0f; }
    if (i < Bv) { nsum[i] = 0.0f; ncnt[i] = 0.0f; }
}

// ---------------------------------------------------------------- kernel 1
// Per-hit: packed argmax (deterministic), per-object energy sums, noise sums.
__global__ void k_hits(const float* __restrict__ beta,
                       const float* __restrict__ pe,
                       const float* __restrict__ te,
                       const int*   __restrict__ tidx,
                       unsigned long long* amax,
                       float* spw, float* spweh, float* nsum, float* ncnt,
                       int N, int BN) {
    int i = blockIdx.x * blockDim.x + threadIdx.x;
    if (i >= BN) return;
    int b = i / N, n = i % N;
    float bc = clipb(beta[i]);
    int t = tidx[i];
    if (t > 0) {
        float a = atanhf(bc);
        float q = a * a + QMIN;                     // q > 0 -> uint bits monotonic
        unsigned long long p =
            ((unsigned long long)__float_as_uint(q) << 32) |
            (unsigned long long)(0xFFFFFFFFu - (unsigned)n);  // tie -> lowest index
        atomicMax(&amax[b * KMAX + t], p);
        float ed = (pe[i] - te[i]) / (te[i] + EDEN);
        float ad = fabsf(ed);
        float eh = (ad <= HUBER) ? 0.5f * ad * ad : HUBER * (ad - 0.5f * HUBER);
        atomicAdd(&spw[b * KMAX + t], bc);
        atomicAdd(&spweh[b * KMAX + t], bc * eh);
    } else {
        atomicAdd(&nsum[b], bc);
        atomicAdd(&ncnt[b], 1.0f);
    }
}

// ---------------------------------------------------------------- kernel 2
// One block per event: decode alpha points, stage WMMA B operands, reduce
// L_beta + L_E + L_noise into one scalar per event.
__global__ void k_obj(const float* __restrict__ beta,
                      const float* __restrict__ cc,
                      const unsigned long long* __restrict__ amax,
                      const float* __restrict__ spw,
                      const float* __restrict__ spweh,
                      const float* __restrict__ nsum,
                      const float* __restrict__ ncnt,
                      float* xax, float* xay, float* xa2, float* qeff,
                      float* scal, int N) {
    int b = blockIdx.x;
    int k = threadIdx.x;
    __shared__ float svf[256], slb[256], sle[256];
    float vf = 0.0f, lb = 0.0f, le = 0.0f;
    if (k < KMAX) {
        unsigned long long p = amax[b * KMAX + k];
        float qx = 0.0f, qy = 0.0f, qa = 0.0f;
        if (p != 0ull) {
            unsigned idx = 0xFFFFFFFFu - (unsigned)(p & 0xFFFFFFFFull);
            qa = __uint_as_float((unsigned)(p >> 32));    // bit-exact q_alpha
            int g = b * N + (int)idx;
            qx = cc[2 * g + 0];
            qy = cc[2 * g + 1];
            float ba = clipb(beta[g]);
            vf = 1.0f;
            lb = 1.0f - ba;
            le = spweh[b * KMAX + k] / (spw[b * KMAX + k] + EPSF);
        }
        xax[b * KMAX + k]  = qx;
        xay[b * KMAX + k]  = qy;
        xa2[b * KMAX + k]  = qx * qx + qy * qy;
        qeff[b * KMAX + k] = qa;                          // 0 for invalid objects
    }
    svf[threadIdx.x] = vf; slb[threadIdx.x] = lb; sle[threadIdx.x] = le;
    __syncthreads();
    for (int s = 128; s > 0; s >>= 1) {
        if (threadIdx.x < s) {
            svf[threadIdx.x] += svf[threadIdx.x + s];
            slb[threadIdx.x] += slb[threadIdx.x + s];
            sle[threadIdx.x] += sle[threadIdx.x + s];
        }
        __syncthreads();
    }
    if (threadIdx.x == 0) {
        float nobj = svf[0] + EPSF;
        scal[b] = slb[0] / nobj + sle[0] / nobj
                + SB * nsum[b] / (ncnt[b] + EPSF);
    }
}

// ---------------------------------------------------------------- kernel 3
// Dense N x K potential pass. Each wave: 16 hits x 192 objects via 12
// V_WMMA_F32_16X16X4_F32 Gram tiles (coords zero-padded to K=4).
__global__ void __launch_bounds__(128)
k_pairs(const float* __restrict__ beta,
        const float* __restrict__ cc,
        const int*   __restrict__ tidx,
        const float* __restrict__ xax, const float* __restrict__ xay,
        const float* __restrict__ xa2, const float* __restrict__ qeff,
        float* part, int N, int nblk) {
    int b    = blockIdx.y;
    int lane = threadIdx.x & 31;
    int wave = threadIdx.x >> 5;
    int tile = blockIdx.x * 4 + wave;            // tile of 16 hits
    int row  = lane & 15;

    unsigned hit = (unsigned)(tile * 16 + row);
    unsigned hc  = hit < (unsigned)N ? hit : (unsigned)(N - 1);
    float vld    = hit < (unsigned)N ? 1.0f : 0.0f;
    int g = b * N + (int)hc;

    float cx = cc[2 * g + 0];
    float cy = cc[2 * g + 1];
    float x2 = cx * cx + cy * cy;
    float qm = qval(beta[g]) * vld;              // q per hit row; 0 kills OOB rows
    int   tm = tidx[g];

    // A-matrix 16x4 f32 layout: lane<16 carries K0,K1 = (x,y); lane>=16 = K2,K3 pad
    v2f A;
    A.x = (lane < 16) ? cx : 0.0f;
    A.y = (lane < 16) ? cy : 0.0f;

    // Pre-broadcast per-row values for the 8 C/D slots (row = v + 8*(lane>=16))
    int half = (lane & 16) >> 1;                 // 0 or 8
    float x2r[8], qr[8];
    int   tr[8];
#pragma unroll
    for (int v = 0; v < 8; ++v) {
        int src = v + half;                      // lanes 0..15 hold rows 0..15
        x2r[v] = __shfl(x2, src, 32);
        qr[v]  = __shfl(qm, src, 32);
        tr[v]  = __shfl(tm, src, 32);
    }

    float acc[8];
#pragma unroll
    for (int v = 0; v < 8; ++v) acc[v] = 0.0f;

    const float* xaxb = xax  + b * KMAX;
    const float* xayb = xay  + b * KMAX;
    const float* xa2b = xa2  + b * KMAX;
    const float* qefb = qeff + b * KMAX;

#pragma unroll 2
    for (int t = 0; t < KMAX / 16; ++t) {
        int k = t * 16 + row;
        float bx = xaxb[k], by = xayb[k];
        v2f Bm;
        Bm.x = (lane < 16) ? bx : 0.0f;          // B 4x16: K rows 0,1 in lanes<16
        Bm.y = (lane < 16) ? by : 0.0f;
        float xa2k = xa2b[k];
        float qk   = qefb[k];                    // 0 for invalid objects

        v8f Cz = {0.f, 0.f, 0.f, 0.f, 0.f, 0.f, 0.f, 0.f};
        v8f D = __builtin_amdgcn_wmma_f32_16x16x4_f32(
            false, A, false, Bm, (short)0, Cz, false, false);

#pragma unroll
        for (int v = 0; v < 8; ++v) {
            float d2   = fmaxf(x2r[v] + xa2k - 2.0f * D[v], 0.0f);
            float att  = d2 * qk;
            // raw v_sqrt_f32: arg = d2+eps >= 1e-6 (no denorm/NaN), ~1ulp is fine
            float dist = __builtin_amdgcn_sqrtf(d2 + EPSF);
            float rep  = fmaxf(0.0f, 1.0f - dist) * qk;
            acc[v] += (tr[v] == k) ? att : rep;
        }
    }

    float s = 0.0f;
#pragma unroll
    for (int v = 0; v < 8; ++v) s += qr[v] * acc[v];
    for (int off = 16; off > 0; off >>= 1) s += __shfl_down(s, off, 32);

    __shared__ float wsum[4];
    if (lane == 0) wsum[wave] = s;
    __syncthreads();
    if (threadIdx.x == 0)
        part[b * nblk + blockIdx.x] = wsum[0] + wsum[1] + wsum[2] + wsum[3];
}

// ---------------------------------------------------------------- kernel 4
__global__ void k_final(const float* __restrict__ part,
                        const float* __restrict__ scal,
                        float* out, int N, int nblk) {
    if (blockIdx.x == 0 && threadIdx.x == 0) {
        float tot = 0.0f;
        for (int b = 0; b < NEVENTS; ++b) {
            float s = 0.0f;
            for (int i = 0; i < nblk; ++i) s += part[b * nblk + i];  // fixed order
            tot += s / (float)N + scal[b];
        }
        out[0] = tot / (float)NEVENTS;
    }
}

// ---------------------------------------------------------------- launcher
extern "C" void kernel_launch(void* const* d_in, const int* in_sizes, int n_in,
                              void* d_out, int out_size, void* d_ws, size_t ws_size,
                              hipStream_t stream) {
    const float* beta = (const float*)d_in[0];
    const float* cc   = (const float*)d_in[1];
    const float* pe   = (const float*)d_in[2];
    const float* te   = (const float*)d_in[3];
    const int*   tidx = (const int*)d_in[4];

    const int Bv = NEVENTS;
    const int BN = in_sizes[0];
    const int N  = BN / Bv;
    const int BK = Bv * KMAX;

    char* w = (char*)d_ws;
    unsigned long long* amax = (unsigned long long*)w; w += (size_t)BK * 8;
    float* spw   = (float*)w; w += (size_t)BK * 4;
    float* spweh = (float*)w; w += (size_t)BK * 4;
    float* nsum  = (float*)w; w += (size_t)Bv * 4;
    float* ncnt  = (float*)w; w += (size_t)Bv * 4;
    float* xax   = (float*)w; w += (size_t)BK * 4;
    float* xay   = (float*)w; w += (size_t)BK * 4;
    float* xa2   = (float*)w; w += (size_t)BK * 4;
    float* qeff  = (float*)w; w += (size_t)BK * 4;
    float* scal  = (float*)w; w += (size_t)Bv * 4;
    float* part  = (float*)w;

    int ntiles = (N + 15) / 16;
    int nblk   = (ntiles + 3) / 4;

    k_init<<<(BK + 255) / 256, 256, 0, stream>>>(amax, spw, spweh, nsum, ncnt, BK, Bv);
    k_hits<<<(BN + 255) / 256, 256, 0, stream>>>(beta, pe, te, tidx, amax,
                                                 spw, spweh, nsum, ncnt, N, BN);
    k_obj<<<Bv, 256, 0, stream>>>(beta, cc, amax, spw, spweh, nsum, ncnt,
                                  xax, xay, xa2, qeff, scal, N);
    dim3 g3(nblk, Bv);
    k_pairs<<<g3, 128, 0, stream>>>(beta, cc, tidx, xax, xay, xa2, qeff,
                                    part, N, nblk);
    k_final<<<1, 32, 0, stream>>>(part, scal, (float*)d_out, N, nblk);
}